// ProbConwayForwardProp_44014824849493
// MI455X (gfx1250) — compile-verified
//
#include <hip/hip_runtime.h>
#include <stdint.h>

// ---------------------------------------------------------------------------
// Probabilistic Conway forward prop, gfx1250.
// 32 x 1024 x 1024 fp32, torus wrap. Bandwidth floor ~11us @ 23.3 TB/s.
// LDS-tiled stencil: async global->LDS halo fill (ASYNCcnt path), packed-fp32
// (v_pk_fma_f32) Poisson-binomial DP, 2 cells/thread-column via float2,
// nontemporal output stores (output is never re-read -> don't pollute L2).
// WMMA intentionally unused: the DP is a per-cell nonlinear recurrence
// (pmf <- M(p_cell) * pmf), not a shared-operand matmul.
// ---------------------------------------------------------------------------

typedef float v2f __attribute__((ext_vector_type(2)));

#define W      1024
#define WMASK  1023
#define TILE   64
#define LCOLS  66            // 64 + 2 halo
#define LROWS  66
#define LSIZE  (LROWS * LCOLS)
#define NTHREADS 256

#if defined(__HIP_DEVICE_COMPILE__) && __has_builtin(__builtin_amdgcn_global_load_async_to_lds_b32)
#define HAS_ASYNC_LDS 1
#else
#define HAS_ASYNC_LDS 0
#endif

// global(AS1) -> LDS(AS3) 4-byte async copy (ASYNCcnt path), with portable
// fallback so the file always compiles on both toolchains.
// Builtin signature (per hipcc diagnostic): (AS1 int*, AS3 int*, imm, imm).
__device__ __forceinline__ void g2lds_b32(const float* g, float* l) {
#if HAS_ASYNC_LDS
  __builtin_amdgcn_global_load_async_to_lds_b32(
      (__attribute__((address_space(1))) int*)(uintptr_t)g,
      (__attribute__((address_space(3))) int*)(uint32_t)(uintptr_t)l,
      /*offset=*/0, /*cpol=*/0);
#else
  *l = *g;
#endif
}

__device__ __forceinline__ void wait_async_done() {
#if HAS_ASYNC_LDS
#if __has_builtin(__builtin_amdgcn_s_wait_asynccnt)
  __builtin_amdgcn_s_wait_asynccnt(0);
#else
  asm volatile("s_wait_asynccnt 0" ::: "memory");
#endif
#endif
}

// One Poisson-binomial DP step for a packed pair of cells.
// pmf_k <- pmf_k*(1-p) + pmf_{k-1}*p, written as 7 packed ops:
//   pmf_k += p*(pmf_{k-1} - pmf_k)   (uses pre-update lower term: eval 3->0)
__device__ __forceinline__ void pmf_step(v2f& p0, v2f& p1, v2f& p2, v2f& p3,
                                         v2f p) {
  p3 = __builtin_elementwise_fma(p, p2 - p3, p3);
  p2 = __builtin_elementwise_fma(p, p1 - p2, p2);
  p1 = __builtin_elementwise_fma(p, p0 - p1, p1);
  p0 = __builtin_elementwise_fma(-p, p0, p0);   // p0 * (1 - p)
}

__global__ __launch_bounds__(NTHREADS)
void ProbConwayForwardProp_kernel(const float* __restrict__ in,
                                  float* __restrict__ out) {
  __shared__ float tile[LSIZE];

  const int z   = blockIdx.z;
  const int ty0 = blockIdx.y * TILE;
  const int tx0 = blockIdx.x * TILE;
  const float* src = in + (size_t)z * (W * W);

  // ---- Halo tile fill: 66x66, torus wrap, async global->LDS ----
  for (int idx = threadIdx.x; idx < LSIZE; idx += NTHREADS) {
    const int r  = idx / LCOLS;
    const int c  = idx - r * LCOLS;
    const int gy = (ty0 + r - 1) & WMASK;
    const int gx = (tx0 + c - 1) & WMASK;
    g2lds_b32(src + ((size_t)gy * W + gx), &tile[idx]);
  }
  wait_async_done();
  __syncthreads();

  // ---- Compute: each thread owns a 2-wide x 8-tall strip ----
  const int tid = threadIdx.x;
  const int txp = tid & 31;        // column-pair index 0..31
  const int wid = tid >> 5;        // row group 0..7 (8 rows each)
  const int lx  = 1 + 2 * txp;     // LDS col of first owned cell
  const int ly0 = 1 + wid * 8;     // LDS row of first owned cell

  // Load LDS row r, cols lx-1..lx+2, as two aligned float2s; build the
  // left/mid/right neighbor pairs for the two owned cells.
  auto loadrow = [&](int r, v2f& L, v2f& M, v2f& R) {
    const float* p = &tile[r * LCOLS + (lx - 1)];  // even index -> 8B aligned
    const v2f A = *(const v2f*)p;        // cols lx-1, lx
    const v2f B = *(const v2f*)(p + 2);  // cols lx+1, lx+2
    L = A;
    M = (v2f){A.y, B.x};
    R = B;
  };

  v2f Lt, Mt, Rt, Lm, Mm, Rm, Lb, Mb, Rb;
  loadrow(ly0 - 1, Lt, Mt, Rt);
  loadrow(ly0,     Lm, Mm, Rm);

  float* dst = out + (size_t)z * (W * W);
  const int gx = tx0 + 2 * txp;

  for (int i = 0; i < 8; ++i) {
    loadrow(ly0 + i + 1, Lb, Mb, Rb);

    v2f p0 = {1.0f, 1.0f};
    v2f p1 = {0.0f, 0.0f};
    v2f p2 = {0.0f, 0.0f};
    v2f p3 = {0.0f, 0.0f};

    pmf_step(p0, p1, p2, p3, Lt);
    pmf_step(p0, p1, p2, p3, Mt);
    pmf_step(p0, p1, p2, p3, Rt);
    pmf_step(p0, p1, p2, p3, Lm);
    pmf_step(p0, p1, p2, p3, Rm);
    pmf_step(p0, p1, p2, p3, Lb);
    pmf_step(p0, p1, p2, p3, Mb);
    pmf_step(p0, p1, p2, p3, Rb);

    // P(live next) = pmf3 + pmf2 * p_self  (self = Mm pair)
    const v2f res = __builtin_elementwise_fma(p2, Mm, p3);

    const int gy = ty0 + wid * 8 + i;
    // Coalesced b64 store, nontemporal: output is write-once, keep it out of
    // L2 so the 192MB L2 holds the input halo lines instead.
    __builtin_nontemporal_store(res, (v2f*)(dst + (size_t)gy * W + gx));

    Lt = Lm; Mt = Mm; Rt = Rm;
    Lm = Lb; Mm = Mb; Rm = Rb;
  }
}

extern "C" void kernel_launch(void* const* d_in, const int* in_sizes, int n_in,
                              void* d_out, int out_size, void* d_ws, size_t ws_size,
                              hipStream_t stream) {
  (void)n_in; (void)out_size; (void)d_ws; (void)ws_size;
  const float* in = (const float*)d_in[0];
  float* out = (float*)d_out;
  const int Z = in_sizes[0] >> 20;  // slices of 1024*1024
  dim3 grid(W / TILE, W / TILE, Z); // 16 x 16 x 32
  ProbConwayForwardProp_kernel<<<grid, dim3(NTHREADS), 0, stream>>>(in, out);
}